// ScaledDotProductAttention_58746562674822
// MI455X (gfx1250) — compile-verified
//
#include <hip/hip_runtime.h>
#include <math.h>

// Problem constants (from reference): B=2, H=16, T=2048, D=64, fp32.
#define B_  2
#define H_  16
#define T_  2048
#define D_  64

#define TILE   16
#define NWAVE  4
#define WAVESZ 32
#define NTHR   (NWAVE * WAVESZ)
#define SROW   (T_ + 4)   // padded LDS row stride (floats): 2052 -> conflict-free strided reads

typedef __attribute__((ext_vector_type(2))) float v2f;
typedef __attribute__((ext_vector_type(8))) float v8f;

// Fused attention for one (b,h) and one 16-row query tile:
//   scores (WMMA f32) -> mask -> softmax (in 128KB LDS) -> attn write -> P@V (WMMA f32) -> out
__global__ __launch_bounds__(NTHR)
void sdpa_fused_kernel(const float* __restrict__ q,
                       const float* __restrict__ k,
                       const float* __restrict__ v,
                       const int*   __restrict__ mask,
                       float* __restrict__ out,
                       float* __restrict__ attn)
{
    __shared__ float S[TILE * SROW];              // 131,328 B : scores / probabilities row block
    __shared__ float outPart[NWAVE * TILE * D_];  //  16,384 B : per-wave partial outputs

    const int bh   = blockIdx.y;          // 0..31  (b*H + h)
    const int t0   = blockIdx.x * TILE;   // query row base
    const int tid  = threadIdx.x;
    const int wave = tid / WAVESZ;
    const int lane = tid % WAVESZ;
    const int lmod = lane & 15;
    const int lhi  = lane >> 4;           // 0 for lanes 0-15, 1 for lanes 16-31

    const float* qb = q + (size_t)bh * T_ * D_;
    const float* kb = k + (size_t)bh * T_ * D_;
    const float* vb = v + (size_t)bh * T_ * D_;

    const float scale = 0.125f;  // 1/sqrt(64), applied to Q like the reference (q*scale)

    // ---- Load Q A-fragments (16x4 f32 chunks), pre-scaled, kept in VGPRs ----
    // A 16x4 layout: lanes 0-15 hold row lmod K={kc*4+0, +1}; lanes 16-31 hold K={+2, +3}.
    v2f aq[D_ / 4];
    {
        const float* qrow = qb + (size_t)(t0 + lmod) * D_;
#pragma unroll
        for (int kc = 0; kc < D_ / 4; ++kc) {
            const int c0 = kc * 4 + lhi * 2;
            aq[kc].x = qrow[c0 + 0] * scale;
            aq[kc].y = qrow[c0 + 1] * scale;
        }
    }

    // ---- Pass 1: S = (Q*scale) @ K^T for this row block, masked, into LDS ----
    // Two independent WMMA accumulator chains (even/odd kc) so the scheduler can
    // interleave them instead of stalling on one serial D->C dependency chain.
    for (int kt = wave; kt < T_ / TILE; kt += NWAVE) {
        const float* krow = kb + (size_t)(kt * TILE + lmod) * D_;
        if (kt + NWAVE < T_ / TILE) {
            // pull next tile's K line toward the WGP while this chain runs
            __builtin_prefetch(kb + (size_t)((kt + NWAVE) * TILE + lmod) * D_, 0, 0);
        }
        v8f c0 = {0.f, 0.f, 0.f, 0.f, 0.f, 0.f, 0.f, 0.f};
        v8f c1 = c0;
#pragma unroll
        for (int kp = 0; kp < D_ / 8; ++kp) {
            const int kcA = 2 * kp;       // even chunk -> chain 0
            const int kcB = 2 * kp + 1;   // odd chunk  -> chain 1
            v2f bA, bB;  // B = K^T chunk: B[d][j] = K[j][d] -> same gather pattern as A
            {
                const int cA = kcA * 4 + lhi * 2;
                bA.x = krow[cA + 0];
                bA.y = krow[cA + 1];
            }
            {
                const int cB = kcB * 4 + lhi * 2;
                bB.x = krow[cB + 0];
                bB.y = krow[cB + 1];
            }
            c0 = __builtin_amdgcn_wmma_f32_16x16x4_f32(
                     false, aq[kcA], false, bA, (short)0, c0, false, false);
            c1 = __builtin_amdgcn_wmma_f32_16x16x4_f32(
                     false, aq[kcB], false, bB, (short)0, c1, false, false);
        }
        const v8f c = c0 + c1;
        // C/D layout: lane<16 -> rows 0..7 in c[0..7]; lane>=16 -> rows 8..15. Col = kt*16+lmod.
        const int col = kt * TILE + lmod;
#pragma unroll
        for (int vr = 0; vr < 8; ++vr) {
            const int rowl = vr + lhi * 8;
            const int m = mask[(size_t)(t0 + rowl) * T_ + col];
            S[rowl * SROW + col] = (m == 0) ? -1e9f : c[vr];
        }
    }
    __syncthreads();

    // ---- Pass 2: row softmax in LDS; stream normalized P to global attn ----
    float* attnb = attn + (size_t)bh * T_ * T_;
    for (int rr = 0; rr < TILE / NWAVE; ++rr) {
        const int rowl = wave * (TILE / NWAVE) + rr;
        float* srow = &S[rowl * SROW];

        float mx = -3.4e38f;
        for (int cdx = lane; cdx < T_; cdx += WAVESZ) mx = fmaxf(mx, srow[cdx]);
#pragma unroll
        for (int off = 16; off > 0; off >>= 1) mx = fmaxf(mx, __shfl_xor(mx, off, WAVESZ));

        float sum = 0.f;
        for (int cdx = lane; cdx < T_; cdx += WAVESZ) {
            const float e = __expf(srow[cdx] - mx);
            srow[cdx] = e;
            sum += e;
        }
#pragma unroll
        for (int off = 16; off > 0; off >>= 1) sum += __shfl_xor(sum, off, WAVESZ);

        const float inv = 1.f / sum;
        float* arow = attnb + (size_t)(t0 + rowl) * T_;
        for (int cdx = lane; cdx < T_; cdx += WAVESZ) {
            const float p = srow[cdx] * inv;
            srow[cdx] = p;       // keep normalized P in LDS for P@V
            arow[cdx] = p;       // coalesced 512MB attn stream-out
        }
    }
    __syncthreads();

    // ---- Pass 3: out = P @ V (contraction over s, split across waves) ----
    // 4 independent accumulator chains (one per 16-wide slice of D).
    v8f acc0 = {0.f,0.f,0.f,0.f,0.f,0.f,0.f,0.f};
    v8f acc1 = acc0, acc2 = acc0, acc3 = acc0;
    for (int st = wave; st < T_ / TILE; st += NWAVE) {
        if (st + NWAVE < T_ / TILE) {
            __builtin_prefetch(vb + (size_t)((st + NWAVE) * TILE + lhi * 2) * D_ + lmod, 0, 0);
        }
#pragma unroll
        for (int kc = 0; kc < 4; ++kc) {
            const int sc = st * TILE + kc * 4 + lhi * 2;
            v2f a;  // A = P chunk from LDS (stride 2052 -> conflict-free)
            a.x = S[lmod * SROW + sc + 0];
            a.y = S[lmod * SROW + sc + 1];
#pragma unroll
            for (int db = 0; db < 4; ++db) {
                const float* vp = vb + (size_t)sc * D_ + db * TILE + lmod;
                v2f b;  // B = V rows sc, sc+1 at column db*16+lmod
                b.x = vp[0];
                b.y = vp[D_];
                v8f* accp = (db == 0) ? &acc0 : (db == 1) ? &acc1 : (db == 2) ? &acc2 : &acc3;
                *accp = __builtin_amdgcn_wmma_f32_16x16x4_f32(
                            false, a, false, b, (short)0, *accp, false, false);
            }
        }
    }
    // Stash per-wave partials
#pragma unroll
    for (int db = 0; db < 4; ++db) {
        const v8f accv = (db == 0) ? acc0 : (db == 1) ? acc1 : (db == 2) ? acc2 : acc3;
#pragma unroll
        for (int vr = 0; vr < 8; ++vr) {
            const int rowl = vr + lhi * 8;
            outPart[(wave * TILE + rowl) * D_ + db * TILE + lmod] = accv[vr];
        }
    }
    __syncthreads();

    // ---- Reduce partials across waves, coalesced write of out tile ----
    float* outb = out + (size_t)bh * T_ * D_ + (size_t)t0 * D_;
    for (int e = tid; e < TILE * D_; e += NTHR) {
        float s = 0.f;
#pragma unroll
        for (int w = 0; w < NWAVE; ++w) s += outPart[w * TILE * D_ + e];
        outb[e] = s;
    }
}

extern "C" void kernel_launch(void* const* d_in, const int* in_sizes, int n_in,
                              void* d_out, int out_size, void* d_ws, size_t ws_size,
                              hipStream_t stream) {
    (void)in_sizes; (void)n_in; (void)out_size; (void)d_ws; (void)ws_size;
    const float* q    = (const float*)d_in[0];
    const float* k    = (const float*)d_in[1];
    const float* v    = (const float*)d_in[2];
    const int*   mask = (const int*)d_in[3];

    // d_out = concat(out[B,H,T,D], attn[B,H,T,T]) in reference return order
    float* out  = (float*)d_out;
    float* attn = out + (size_t)B_ * H_ * T_ * D_;

    dim3 grid(T_ / TILE, B_ * H_);
    dim3 block(NTHR);
    sdpa_fused_kernel<<<grid, block, 0, stream>>>(q, k, v, mask, out, attn);
}